// MAB_13589276524917
// MI455X (gfx1250) — compile-verified
//
#include <hip/hip_runtime.h>
#include <hip/hip_bf16.h>

typedef __attribute__((ext_vector_type(2))) float v2f;
typedef __attribute__((ext_vector_type(8))) float v8f;

#define CH 16
#define TT 31
#define HWD 65536
#define CHS (TT * HWD)            // channel stride in elements: 2031616
#define NT 2                      // position-tiles per wave
#define POS_PER_BLOCK (8 * NT * 16)        // 256
#define BLOCKS_PER_B (HWD / POS_PER_BLOCK) // 256

static __device__ __forceinline__ v8f wmma_f32(v2f a, v2f b, v8f c) {
  // D = A(16x4 f32) * B(4x16 f32) + C(16x16 f32)
  return __builtin_amdgcn_wmma_f32_16x16x4_f32(
      /*neg_a=*/false, a, /*neg_b=*/false, b,
      /*c_mod=*/(short)0, c, /*reuse_a=*/false, /*reuse_b=*/false);
}

static __device__ __forceinline__ float fast_tanh(float x) {
#if __has_builtin(__builtin_amdgcn_tanhf)
  return __builtin_amdgcn_tanhf(x);       // v_tanh_f32 (confirmed in round 1 asm)
#else
  return tanhf(x);
#endif
}

static __device__ __forceinline__ float fast_sigmoid(float x) {
  return __builtin_amdgcn_rcpf(1.0f + __expf(-x)); // v_exp_f32 + v_rcp_f32
}

// xor-16 half-wave swap. Prefer v_permlanex16_b32 (VALU pipe) over
// __shfl_xor's ds_bpermute lowering (shared LDS pipe).
static __device__ __forceinline__ float swap16(float v) {
#if __has_builtin(__builtin_amdgcn_permlanex16)
  // Identity nibble selectors: lane i exchanges with lane i^16.
  int iv = __float_as_int(v);
  return __int_as_float(__builtin_amdgcn_permlanex16(
      iv, iv, 0x76543210, 0xFEDCBA98, false, false));
#else
  return __shfl_xor(v, 16, 32);
#endif
}

// Convert a 16x16 result in C/D layout (8 VGPRs; lanes 0-15: ch 0-7 of pos n,
// lanes 16-31: ch 8-15 of pos n) into the B-operand layout for the next
// matmul: slice kk VGPR j holds channel 4*kk + j + 2*half at position n.
static __device__ __forceinline__ void c2b(const v8f d, v2f bb[4], int half) {
  float s[8];
#pragma unroll
  for (int r = 0; r < 8; ++r) s[r] = swap16(d[r]);
  bb[0].x = half ? s[2] : d[0];  bb[0].y = half ? s[3] : d[1];
  bb[1].x = half ? s[6] : d[4];  bb[1].y = half ? s[7] : d[5];
  bb[2].x = half ? d[2] : s[0];  bb[2].y = half ? d[3] : s[1];
  bb[3].x = half ? d[6] : s[4];  bb[3].y = half ? d[7] : s[5];
}

// Pass 1: fused gates + time-scan. Reads x once (double-buffered one timestep
// ahead), keeps h in registers across t, writes unscaled h to out, emits
// per-block channel partial sums (deterministic, no atomics).
__global__ __launch_bounds__(256) void k_fused(
    const float* __restrict__ x,
    const float* __restrict__ wf1, const float* __restrict__ bf1,
    const float* __restrict__ wf2, const float* __restrict__ bf2,
    const float* __restrict__ ww1, const float* __restrict__ bw1,
    const float* __restrict__ ww2, const float* __restrict__ bw2,
    float* __restrict__ out, float* __restrict__ part) {
  const int lane = threadIdx.x & 31;
  const int wave = threadIdx.x >> 5;
  const int half = lane >> 4;
  const int n    = lane & 15;
  const int b    = blockIdx.x / BLOCKS_PER_B;
  const int hw0  = (blockIdx.x % BLOCKS_PER_B) * POS_PER_BLOCK;

  // Weights in A layout: lane (m = n, half) holds W[m, 4kk+2*half + {0,1}].
  v2f Af1[4], Af2[4], Aw1[4], Aw2[4];
#pragma unroll
  for (int kk = 0; kk < 4; ++kk) {
    const int col = 4 * kk + 2 * half;
    Af1[kk].x = wf1[n * CH + col]; Af1[kk].y = wf1[n * CH + col + 1];
    Af2[kk].x = wf2[n * CH + col]; Af2[kk].y = wf2[n * CH + col + 1];
    Aw1[kk].x = ww1[n * CH + col]; Aw1[kk].y = ww1[n * CH + col + 1];
    Aw2[kk].x = ww2[n * CH + col]; Aw2[kk].y = ww2[n * CH + col + 1];
  }
  // Biases broadcast in C layout (reg r = bias[r + 8*half], all positions).
  v8f Cf1, Cf2, Cw1, Cw2;
#pragma unroll
  for (int r = 0; r < 8; ++r) {
    Cf1[r] = bf1[r + 8 * half];
    Cf2[r] = bf2[r + 8 * half];
    Cw1[r] = bw1[r + 8 * half];
    Cw2[r] = bw2[r + 8 * half];
  }

  v8f h[NT];
  float sacc[8];
#pragma unroll
  for (int j = 0; j < NT; ++j)
#pragma unroll
    for (int r = 0; r < 8; ++r) h[j][r] = 0.0f;
#pragma unroll
  for (int r = 0; r < 8; ++r) sacc[r] = 0.0f;

  const size_t bbase = (size_t)b * CH * CHS;
  // Per-lane base offset within one (t) slab: position only (channel offset added per load).
  const int pbase[NT] = { hw0 + (wave * NT + 0) * 16 + n,
                          hw0 + (wave * NT + 1) * 16 + n };

  // Prologue: load tiles for t = 0 in B layout.
  v2f bx[NT][4];
#pragma unroll
  for (int j = 0; j < NT; ++j) {
    const float* xb = x + bbase + pbase[j];
#pragma unroll
    for (int kk = 0; kk < 4; ++kk) {
      const int c = 4 * kk + 2 * half;
      bx[j][kk].x = xb[(size_t)c * CHS];
      bx[j][kk].y = xb[(size_t)(c + 1) * CHS];
    }
  }

  for (int t = 0; t < TT; ++t) {
    // Issue next timestep's loads first so the WMMA/VALU phase hides them.
    v2f bxn[NT][4];
    if (t + 1 < TT) {
#pragma unroll
      for (int j = 0; j < NT; ++j) {
        const float* xb = x + bbase + (size_t)(t + 1) * HWD + pbase[j];
#pragma unroll
        for (int kk = 0; kk < 4; ++kk) {
          const int c = 4 * kk + 2 * half;
          bxn[j][kk].x = xb[(size_t)c * CHS];
          bxn[j][kk].y = xb[(size_t)(c + 1) * CHS];
        }
      }
    }

#pragma unroll
    for (int j = 0; j < NT; ++j) {
      // Layer 1 of both FFNs (independent accumulators -> ILP).
      v8f z = Cf1, f = Cw1;
#pragma unroll
      for (int kk = 0; kk < 4; ++kk) z = wmma_f32(Af1[kk], bx[j][kk], z);
#pragma unroll
      for (int kk = 0; kk < 4; ++kk) f = wmma_f32(Aw1[kk], bx[j][kk], f);
#pragma unroll
      for (int r = 0; r < 8; ++r) { z[r] = fast_tanh(z[r]); f[r] = fast_tanh(f[r]); }
      v2f bz[4], bfv[4];
      c2b(z, bz, half);
      c2b(f, bfv, half);
      // Layer 2 of both FFNs.
      v8f z2 = Cf2, f2 = Cw2;
#pragma unroll
      for (int kk = 0; kk < 4; ++kk) z2 = wmma_f32(Af2[kk], bz[kk], z2);
#pragma unroll
      for (int kk = 0; kk < 4; ++kk) f2 = wmma_f32(Aw2[kk], bfv[kk], f2);
      // Gate activations + recurrence + store + running channel sums.
      float* ob = out + bbase + (size_t)t * HWD + pbase[j];
#pragma unroll
      for (int r = 0; r < 8; ++r) {
        const float zz = fast_tanh(z2[r]);
        const float ff = fast_sigmoid(f2[r]);
        const float hn = ff * (h[j][r] - zz) + zz;  // f*h + (1-f)*z
        h[j][r] = hn;
        sacc[r] += hn;
        ob[(size_t)(r + 8 * half) * CHS] = hn;
      }
    }

    // Rotate double buffer.
#pragma unroll
    for (int j = 0; j < NT; ++j)
#pragma unroll
      for (int kk = 0; kk < 4; ++kk) bx[j][kk] = bxn[j][kk];
  }

  // Reduce over the 16 positions within each half-wave (channels stay fixed).
#pragma unroll
  for (int m = 1; m < 16; m <<= 1)
#pragma unroll
    for (int r = 0; r < 8; ++r) sacc[r] += __shfl_xor(sacc[r], m, 32);

  __shared__ float lsum[8][CH];
  if (n == 0) {
#pragma unroll
    for (int r = 0; r < 8; ++r) lsum[wave][r + 8 * half] = sacc[r];
  }
  __syncthreads();
  if (threadIdx.x < CH) {
    float s = 0.0f;
#pragma unroll
    for (int w = 0; w < 8; ++w) s += lsum[w][threadIdx.x];
    part[blockIdx.x * CH + threadIdx.x] = s;
  }
}

// Pass 2: reduce block partials -> mean -> 16x16 matvec -> sigmoid. 32 threads.
__global__ void k_att(const float* __restrict__ part,
                      const float* __restrict__ wsca,
                      const float* __restrict__ bsca,
                      float* __restrict__ att) {
  __shared__ float mean[2 * CH];
  const int tid = threadIdx.x;              // 0..31
  const int b = tid >> 4, c = tid & 15;
  float s = 0.0f;
  for (int blk = 0; blk < BLOCKS_PER_B; ++blk)
    s += part[(b * BLOCKS_PER_B + blk) * CH + c];
  mean[tid] = s * (1.0f / ((float)TT * (float)HWD));
  __syncthreads();
  float a = bsca[c];
#pragma unroll
  for (int cc = 0; cc < CH; ++cc) a += wsca[c * CH + cc] * mean[b * CH + cc];
  att[tid] = fast_sigmoid(a);
}

// Pass 3: streaming scale of out by att[b*16+c], float4 vectorized.
__global__ __launch_bounds__(256) void k_scale(float* __restrict__ out,
                                               const float* __restrict__ att) {
  const int i = blockIdx.x * 256 + threadIdx.x;   // float4 index
  float4* o4 = (float4*)out;
  float4 v = o4[i];
  const int bc = (i * 4) / CHS;                   // (b*16 + c), fits in int32
  const float a = att[bc];
  v.x *= a; v.y *= a; v.z *= a; v.w *= a;
  o4[i] = v;
}

extern "C" void kernel_launch(void* const* d_in, const int* in_sizes, int n_in,
                              void* d_out, int out_size, void* d_ws, size_t ws_size,
                              hipStream_t stream) {
  const float* x    = (const float*)d_in[0];
  const float* wf1  = (const float*)d_in[1];
  const float* bf1  = (const float*)d_in[2];
  const float* wf2  = (const float*)d_in[3];
  const float* bf2  = (const float*)d_in[4];
  const float* ww1  = (const float*)d_in[5];
  const float* bw1  = (const float*)d_in[6];
  const float* ww2  = (const float*)d_in[7];
  const float* bw2  = (const float*)d_in[8];
  const float* wsca = (const float*)d_in[9];
  const float* bsca = (const float*)d_in[10];
  float* out  = (float*)d_out;
  float* part = (float*)d_ws;                 // 512*16 floats = 32 KB
  float* att  = part + 2 * BLOCKS_PER_B * CH; // 32 floats

  k_fused<<<2 * BLOCKS_PER_B, 256, 0, stream>>>(
      x, wf1, bf1, wf2, bf2, ww1, bw1, ww2, bw2, out, part);
  k_att<<<1, 32, 0, stream>>>(part, wsca, bsca, att);
  const int n4 = (2 * CH * TT * HWD) / 4;     // 16,252,928 float4s
  k_scale<<<n4 / 256, 256, 0, stream>>>(out, att);
}